// SparseLinear2_59339268161791
// MI455X (gfx1250) — compile-verified
//
#include <hip/hip_runtime.h>
#include <cstdint>

// ---------------------------------------------------------------------------
// SpMM (COO, E=1.6M nnz) x (N x 32 dense) -> (M x 32), memory-bound.
// wave32 trick: lane == batch column. One edge => one 128B coalesced gather
// + one 128B coalesced global_atomic_add_f32. Edge stream staged into LDS
// with gfx1250 async-to-LDS copies (ASYNCcnt).
// ---------------------------------------------------------------------------

#define TILE_E 1024
#define BCOLS  32

__device__ __forceinline__ uint32_t lds_off_of(const void* p) {
    // ISA 10.2: for LDS aperture, addr[31:0] IS the LDS byte offset.
    return (uint32_t)(uintptr_t)p;
}

__device__ __forceinline__ void async_copy_b128(uint32_t lds_byte_off,
                                                const void* sbase_uniform,
                                                uint32_t vgpr_byte_off) {
    // GVS mode: mem = SADDR(64) + VADDR(32) ; LDS dest offset in first VGPR.
    asm volatile("global_load_async_to_lds_b128 %0, %1, %2"
                 :
                 : "v"(lds_byte_off), "v"(vgpr_byte_off), "s"(sbase_uniform)
                 : "memory");
}

__device__ __forceinline__ void wait_async0() {
    asm volatile("s_wait_asynccnt 0" ::: "memory");
}

// ---------------------------------------------------------------------------
// Kernel 1: x (32 x N, row-major) -> x_t (N x 32), and zero out_t (M x 32).
// 32x32 LDS tile transpose, block = (32, 8).
// ---------------------------------------------------------------------------
__global__ void __launch_bounds__(256) k_init(const float* __restrict__ x,
                                              float* __restrict__ x_t,
                                              float* __restrict__ out_t,
                                              int N_, int M_) {
    __shared__ float tile[32][33];
    const int tx = threadIdx.x;          // 0..31
    const int ty = threadIdx.y;          // 0..7
    const int n0 = blockIdx.x * 32;

#pragma unroll
    for (int i = 0; i < 4; ++i) {
        const int b = ty + i * 8;        // batch row 0..31
        const int n = n0 + tx;
        tile[b][tx] = (n < N_) ? x[(size_t)b * N_ + n] : 0.0f;
    }
    __syncthreads();
#pragma unroll
    for (int i = 0; i < 4; ++i) {
        const int nl = ty + i * 8;
        const int n  = n0 + nl;
        if (n < N_) x_t[(size_t)n * BCOLS + tx] = tile[tx][nl];
    }
    // zero accumulator tile (M == N sized grid covers both; guard anyway)
#pragma unroll
    for (int i = 0; i < 4; ++i) {
        const int ml = ty + i * 8;
        const int m  = n0 + ml;
        if (m < M_) out_t[(size_t)m * BCOLS + tx] = 0.0f;
    }
}

// ---------------------------------------------------------------------------
// Kernel 2: edge scatter. One block = one tile of TILE_E edges.
// Stage src64/dst64/val into LDS via async b128 copies, then 8 waves each
// process a contiguous 128-edge slice; lane = batch column.
// ---------------------------------------------------------------------------
__global__ void __launch_bounds__(256) k_edges(const uint32_t* __restrict__ idx, // 2*E int64 (as u32 pairs)
                                               const float* __restrict__ values,
                                               const float* __restrict__ x_t,
                                               float* __restrict__ out_t,
                                               int E_) {
    __shared__ __align__(16) unsigned char smem[TILE_E * 8 + TILE_E * 8 + TILE_E * 4];

    const int tid = threadIdx.x;
    const int e0  = blockIdx.x * TILE_E;
    const int cnt = (E_ - e0 < TILE_E) ? (E_ - e0) : TILE_E;

    const uint32_t lds0 = lds_off_of(&smem[0]);
    const char* src_g = (const char*)idx + (size_t)e0 * 8;
    const char* dst_g = (const char*)idx + (size_t)E_ * 8 + (size_t)e0 * 8;
    const char* val_g = (const char*)values + (size_t)e0 * 4;

    // src64: TILE_E*8 bytes = 512 x 16B chunks (2 edges per chunk)
    {
        uint32_t c = (uint32_t)tid;
        if ((int)(c * 2) < cnt) async_copy_b128(lds0 + c * 16, src_g, c * 16);
        c += 256;
        if ((int)(c * 2) < cnt) async_copy_b128(lds0 + c * 16, src_g, c * 16);
    }
    // dst64: same shape, at LDS offset TILE_E*8
    {
        uint32_t c = (uint32_t)tid;
        if ((int)(c * 2) < cnt) async_copy_b128(lds0 + TILE_E * 8 + c * 16, dst_g, c * 16);
        c += 256;
        if ((int)(c * 2) < cnt) async_copy_b128(lds0 + TILE_E * 8 + c * 16, dst_g, c * 16);
    }
    // values: TILE_E*4 bytes = 256 x 16B chunks (4 edges per chunk)
    {
        uint32_t c = (uint32_t)tid;
        if ((int)(c * 4) < cnt) async_copy_b128(lds0 + TILE_E * 16 + c * 16, val_g, c * 16);
    }
    wait_async0();        // per-wave ASYNCcnt == 0
    __syncthreads();      // all waves' copies visible

    const uint32_t* s_src = (const uint32_t*)(smem);
    const uint32_t* s_dst = (const uint32_t*)(smem + TILE_E * 8);
    const float*    s_val = (const float*)(smem + TILE_E * 16);

    const int lane = tid & 31;           // == batch column b
    const int wv   = tid >> 5;           // wave id 0..7
    const int beg  = wv * (TILE_E / 8);
    const int end  = (beg + TILE_E / 8 < cnt) ? (beg + TILE_E / 8) : cnt;

    for (int le = beg; le < end; ++le) {
        // uniform-address LDS broadcasts (low word of int64 index suffices)
        const uint32_t s = s_src[2 * le];
        const uint32_t d = s_dst[2 * le];
        const float    v = s_val[le];
        const float   xv = x_t[(size_t)s * BCOLS + lane];     // 128B coalesced gather (L2-resident)
        atomicAdd(&out_t[(size_t)d * BCOLS + lane], v * xv);  // 128B coalesced f32 atomic, no return
    }
}

// ---------------------------------------------------------------------------
// Kernel 3: out_t (M x 32) -> out (32 x M) with bias. 32x32 LDS tile.
// ---------------------------------------------------------------------------
__global__ void __launch_bounds__(256) k_final(const float* __restrict__ out_t,
                                               const float* __restrict__ bias,
                                               float* __restrict__ out,
                                               int M_) {
    __shared__ float tile[32][33];
    const int tx = threadIdx.x;
    const int ty = threadIdx.y;
    const int m0 = blockIdx.x * 32;

#pragma unroll
    for (int i = 0; i < 4; ++i) {
        const int ml = ty + i * 8;
        const int m  = m0 + ml;
        tile[ml][tx] = (m < M_) ? out_t[(size_t)m * BCOLS + tx] : 0.0f;
    }
    __syncthreads();
#pragma unroll
    for (int i = 0; i < 4; ++i) {
        const int b = ty + i * 8;
        const int m = m0 + tx;
        if (m < M_) out[(size_t)b * M_ + m] = tile[tx][b] + bias[m];
    }
}

// ---------------------------------------------------------------------------
extern "C" void kernel_launch(void* const* d_in, const int* in_sizes, int n_in,
                              void* d_out, int out_size, void* d_ws, size_t ws_size,
                              hipStream_t stream) {
    const float*    x      = (const float*)d_in[0];     // (B, N, 1) f32
    const float*    values = (const float*)d_in[1];     // (E,)      f32
    const float*    bias   = (const float*)d_in[2];     // (M, 1)    f32
    const uint32_t* idx    = (const uint32_t*)d_in[3];  // (2, E)    int64 -> u32 pairs

    const int B_ = 32;
    const int N_ = in_sizes[0] / B_;
    const int E_ = in_sizes[1];
    const int M_ = in_sizes[2];

    float* x_t   = (float*)d_ws;                        // N*32 floats (6.4 MB)
    float* out_t = x_t + (size_t)N_ * BCOLS;            // M*32 floats (6.4 MB)
    float* out   = (float*)d_out;

    const dim3 blk(32, 8);
    const int  nmax   = (N_ > M_) ? N_ : M_;
    const int  tilesI = (nmax + 31) / 32;
    const int  tilesE = (E_ + TILE_E - 1) / TILE_E;
    const int  tilesM = (M_ + 31) / 32;

    k_init <<<tilesI, blk, 0, stream>>>(x, x_t, out_t, N_, M_);
    k_edges<<<tilesE, 256, 0, stream>>>(idx, values, x_t, out_t, E_);
    k_final<<<tilesM, blk, 0, stream>>>(out_t, bias, out, M_);
}